// interpolate_sst_42314017800439
// MI455X (gfx1250) — compile-verified
//
#include <hip/hip_runtime.h>
#include <hip/hip_bf16.h>
#include <stdint.h>

#define LON   1440
#define HGT   721
#define BATCH 32
#define KSTEP 45            // LON / 32
#define NROW  (BATCH * HGT) // 23072
#define WPB   4             // waves (rows) per block

__global__ __launch_bounds__(WPB * 32)
void k_interp_row(const float* __restrict__ sst,
                  const int*   __restrict__ lsm,
                  float*       __restrict__ out,
                  int*         __restrict__ flags) {
  __shared__ __align__(16) float s_sst[WPB][LON];
  __shared__ short               s_pidx[WPB][LON];
  __shared__ unsigned int        s_mb[WPB][KSTEP + 3];

  const int lane = threadIdx.x & 31;
  const int w    = threadIdx.x >> 5;
  const int row  = blockIdx.x * WPB + w;          // grid is exact: NROW % WPB == 0

  const float* grow = sst + (size_t)row * LON;
  const int*   lrow = lsm + (size_t)row * LON;
  float*       orow = out + (size_t)row * LON;

  // ---- Stage the sst row into LDS via CDNA5 async global->LDS DMA (ASYNCcnt) ----
  {
    unsigned long long ga = (unsigned long long)(uintptr_t)grow + (unsigned)lane * 16u;
    unsigned int       la = (unsigned int)(uintptr_t)&s_sst[w][0] + (unsigned)lane * 16u;
#pragma unroll
    for (int j = 0; j < 11; ++j) {   // 11 * 512B = 5632B
      asm volatile("global_load_async_to_lds_b128 %0, %1, off"
                   :: "v"(la), "v"(ga) : "memory");
      ga += 512; la += 512;
    }
    // tail 128B (floats 1408..1439)
    unsigned long long gt = (unsigned long long)(uintptr_t)grow + 5632u + (unsigned)lane * 4u;
    unsigned int       lt = (unsigned int)(uintptr_t)&s_sst[w][0] + 5632u + (unsigned)lane * 4u;
    asm volatile("global_load_async_to_lds_b32 %0, %1, off"
                 :: "v"(lt), "v"(gt) : "memory");
    asm volatile("s_wait_asynccnt 0" ::: "memory");
  }

  // ---- Forward sweep: prev = inclusive max-scan of (m ? i : -1) ----
  int carry = -1;
#pragma unroll 1
  for (int k = 0; k < KSTEP; ++k) {
    const int e = (k << 5) + lane;
    const int lv = lrow[e];
    const float sv = s_sst[w][e];
    const bool m = (lv == 0) && (sv == sv);
    const unsigned int mb = (unsigned int)__ballot(m);
    if (lane == 0) s_mb[w][k] = mb;
    int x = m ? e : -1;
#pragma unroll
    for (int d = 1; d < 32; d <<= 1) {
      const int y = __shfl_up(x, d, 32);
      if (lane >= d) x = x > y ? x : y;
    }
    x = x > carry ? x : carry;
    s_pidx[w][e] = (short)x;
    carry = __shfl(x, 31, 32);
  }
  const int  rowLast = carry;
  const bool hv      = rowLast >= 0;
  if (lane == 0) flags[row] = hv ? 1 : 0;

  // ---- rowFirst from the validity bitmask (ffs + xor-reduce) ----
  int rowFirst;
  {
    const unsigned int b1 = s_mb[w][lane];
    const int c2 = lane + 32;
    const unsigned int b2 = (c2 < KSTEP) ? s_mb[w][c2] : 0u;
    int f1 = b1 ? ((lane << 5) + (__ffs(b1) - 1)) : (2 * LON);
    int f2 = b2 ? ((c2 << 5) + (__ffs(b2) - 1)) : (2 * LON);
    int f = f1 < f2 ? f1 : f2;
#pragma unroll
    for (int d = 16; d > 0; d >>= 1) {
      const int y = __shfl_xor(f, d, 32);
      f = f < y ? f : y;
    }
    rowFirst = f;
  }

  // ---- Backward sweep fused with interpolation + store ----
  const float qnan = __builtin_nanf("");
  int carryN = 2 * LON;
#pragma unroll 1
  for (int k = KSTEP - 1; k >= 0; --k) {
    const int e = (k << 5) + lane;
    const unsigned int mb = s_mb[w][k];
    const bool m = (mb >> lane) & 1u;
    int x = m ? e : 2 * LON;
#pragma unroll
    for (int d = 1; d < 32; d <<= 1) {
      const int y = __shfl_down(x, d, 32);
      if (lane + d < 32) x = x < y ? x : y;
    }
    x = x < carryN ? x : carryN;     // include carry from higher-k groups
    carryN = __shfl(x, 0, 32);

    const int   prevRaw = (int)s_pidx[w][e];
    const float sv      = s_sst[w][e];

    const int pI = (prevRaw >= 0) ? prevRaw : rowLast - LON;
    const int pG = (prevRaw >= 0) ? prevRaw : (hv ? rowLast : 0);
    const int nI = (x < 2 * LON)  ? x       : rowFirst + LON;
    const int nG = (x < 2 * LON)  ? x       : (hv ? rowFirst : 0);

    const float pV = s_sst[w][pG];
    const float nV = s_sst[w][nG];

    const int dtot = nI - pI;
    const int dp   = e - pI;
    const int dden = dtot > 1 ? dtot : 1;
    const float t  = (dtot > 0) ? ((float)dp / (float)dden) : 0.0f;
    float v = pV * (1.0f - t) + nV * t;
    if (!hv) v = qnan;

    orow[e] = (sv == sv) ? sv : v;
  }
}

__global__ __launch_bounds__(128)
void k_lastrow(const int* __restrict__ flags, int* __restrict__ lastrow) {
  __shared__ int red[128];
  const int b = blockIdx.x;
  int best = -1;
  for (int h = threadIdx.x; h < HGT; h += 128)
    if (flags[b * HGT + h] && h > best) best = h;
  red[threadIdx.x] = best;
  __syncthreads();
  for (int s = 64; s > 0; s >>= 1) {
    if ((int)threadIdx.x < s) {
      const int o = red[threadIdx.x + s];
      if (o > red[threadIdx.x]) red[threadIdx.x] = o;
    }
    __syncthreads();
  }
  if (threadIdx.x == 0) lastrow[b] = red[0];
}

__global__ __launch_bounds__(256)
void k_tailfill(const float* __restrict__ sst,
                float*       __restrict__ out,
                const int*   __restrict__ lastrow) {
  const int rid = blockIdx.x;
  const int b = rid / HGT;
  const int h = rid % HGT;
  const int lr = lastrow[b];
  if (lr < 0 || h <= lr) return;
  const float* srow = sst + (size_t)rid * LON;
  const float* frow = out + ((size_t)b * HGT + lr) * LON;   // == v at lastrow
  float*       orow = out + (size_t)rid * LON;
  for (int i = threadIdx.x; i < LON; i += 256) {
    const float s = srow[i];
    orow[i] = (s == s) ? s : frow[i];
  }
}

extern "C" void kernel_launch(void* const* d_in, const int* in_sizes, int n_in,
                              void* d_out, int out_size, void* d_ws, size_t ws_size,
                              hipStream_t stream) {
  const float* sst = (const float*)d_in[0];
  const int*   lsm = (const int*)d_in[1];
  float*       out = (float*)d_out;

  int* flags   = (int*)d_ws;           // NROW ints
  int* lastrow = flags + NROW;         // BATCH ints

  k_interp_row<<<NROW / WPB, WPB * 32, 0, stream>>>(sst, lsm, out, flags);
  k_lastrow<<<BATCH, 128, 0, stream>>>(flags, lastrow);
  k_tailfill<<<NROW, 256, 0, stream>>>(sst, out, lastrow);
}